// SAGEExpert_2310692405502
// MI455X (gfx1250) — compile-verified
//
#include <hip/hip_runtime.h>

// ---------------------------------------------------------------------------
// GraphSAGE 2-layer forward for MI455X (gfx1250, wave32).
// Edge scatter is L2-resident (51MB feature table << 192MB L2); GEMMs are
// ~10 GFLOP total -> use fp32 WMMA (v_wmma_f32_16x16x4_f32) with fused
// mean-div / bias / BN-stat epilogues.
// ---------------------------------------------------------------------------

#define N_NODES 100000
#define N_EDGES 3200000
#define IN_DIM  128
#define HID_DIM 128
#define OUT_DIM 64
#define BN_EPS  1e-5f

typedef __attribute__((ext_vector_type(2))) float v2f;
typedef __attribute__((ext_vector_type(8))) float v8f;

// ---------------- zero workspace (re-run every call: atomics accumulate) ----
__global__ void k_zero(float* __restrict__ p, long n) {
  long i = (long)blockIdx.x * blockDim.x + threadIdx.x;
  long stride = (long)gridDim.x * blockDim.x;
  for (; i < n; i += stride) p[i] = 0.0f;
}

// ---------------- in-degree counts -----------------------------------------
__global__ void k_count(const int* __restrict__ ei, float* __restrict__ cnt) {
  int e = blockIdx.x * blockDim.x + threadIdx.x;
  if (e < N_EDGES) atomicAdd(&cnt[ei[N_EDGES + e]], 1.0f);
}

// ---------------- scatter-add feat[src] into agg[dst] (128 channels) -------
// one work item = (edge, 4-channel group); 32 groups per edge
__global__ void k_scatter(const int* __restrict__ ei,
                          const float* __restrict__ feat,
                          float* __restrict__ agg) {
  long idx = (long)blockIdx.x * blockDim.x + threadIdx.x;
  const long total = (long)N_EDGES * 32;
  if (idx >= total) return;
  int e = (int)(idx >> 5);
  int g = (int)(idx & 31);
  int src = ei[e];
  int dst = ei[N_EDGES + e];
  const float4 v = *(const float4*)(feat + (long)src * 128 + g * 4);
  float* d = agg + (long)dst * 128 + g * 4;
  atomicAdd(d + 0, v.x);
  atomicAdd(d + 1, v.y);
  atomicAdd(d + 2, v.z);
  atomicAdd(d + 3, v.w);
}

// ---------------- layer 1 GEMM: (agg/cnt)@Wl + x@Wr + b, fused BN sums -----
// one wave per 16-row tile, full N=128 (8 accumulator tiles), K=128 in 4-steps
__global__ void k_gemm1(const float* __restrict__ agg, const float* __restrict__ cnt,
                        const float* __restrict__ x,
                        const float* __restrict__ Wl, const float* __restrict__ Wr,
                        const float* __restrict__ bias,
                        float* __restrict__ hpre,
                        float* __restrict__ bnSum, float* __restrict__ bnSq) {
  const int wave = blockIdx.x * (blockDim.x >> 5) + (threadIdx.x >> 5);
  if (wave >= N_NODES / 16) return;     // whole-wave exit: EXEC stays all-1s
  const int lane  = threadIdx.x & 31;
  const int laneM = lane & 15;          // M for A-frag / N for B-frag
  const int kHalf = lane >> 4;          // 0: K={0,1}, 1: K={2,3}
  const int m0 = wave * 16;
  const int rowA = m0 + laneM;

  const float rInv = 1.0f / fmaxf(cnt[rowA], 1.0f);
  const float* aggRow = agg + (long)rowA * 128;
  const float* xRow   = x   + (long)rowA * 128;

  v8f acc[8] = {};
  for (int k0 = 0; k0 < 128; k0 += 4) {
    const int kb = k0 + 2 * kHalf;
    v2f a1, a2;
    a1.x = aggRow[kb] * rInv;  a1.y = aggRow[kb + 1] * rInv;
    a2.x = xRow[kb];           a2.y = xRow[kb + 1];
#pragma unroll
    for (int t = 0; t < 8; ++t) {
      const int n = t * 16 + laneM;
      v2f bl, br;
      bl.x = Wl[kb * 128 + n];        bl.y = Wl[(kb + 1) * 128 + n];
      br.x = Wr[kb * 128 + n];        br.y = Wr[(kb + 1) * 128 + n];
      acc[t] = __builtin_amdgcn_wmma_f32_16x16x4_f32(false, a1, false, bl,
                                                     (short)0, acc[t], false, false);
      acc[t] = __builtin_amdgcn_wmma_f32_16x16x4_f32(false, a2, false, br,
                                                     (short)0, acc[t], false, false);
    }
  }

  // epilogue: +bias, store pre-BN activations, per-channel sum / sumsq atomics
#pragma unroll
  for (int t = 0; t < 8; ++t) {
    const int n = t * 16 + laneM;
    const float bv = bias[n];
    float s = 0.0f, s2 = 0.0f;
#pragma unroll
    for (int i = 0; i < 8; ++i) {
      // C/D layout: VGPR i, lanes 0-15 -> M=i, lanes 16-31 -> M=i+8, N=lane&15
      const float v = acc[t][i] + bv;
      const int row = m0 + i + (kHalf ? 8 : 0);
      hpre[(long)row * 128 + n] = v;
      s += v;
      s2 += v * v;
    }
    atomicAdd(&bnSum[n], s);
    atomicAdd(&bnSq[n], s2);
  }
}

// ---------------- BN finalize: scale/shift from batch statistics -----------
__global__ void k_bnfin(const float* __restrict__ bnSum, const float* __restrict__ bnSq,
                        const float* __restrict__ gamma, const float* __restrict__ beta,
                        float* __restrict__ scale, float* __restrict__ shift) {
  int c = threadIdx.x;
  if (c < HID_DIM) {
    const float inv_n = 1.0f / (float)N_NODES;
    float mu  = bnSum[c] * inv_n;
    float var = bnSq[c] * inv_n - mu * mu;           // biased variance
    float sc  = gamma[c] * rsqrtf(var + BN_EPS);
    scale[c] = sc;
    shift[c] = beta[c] - mu * sc;
  }
}

// ---------------- apply BN+ReLU in place; zero layer-2 accumulator ---------
__global__ void k_bnrelu_zero(float* __restrict__ h, float* __restrict__ agg2,
                              const float* __restrict__ scale,
                              const float* __restrict__ shift) {
  long i = (long)blockIdx.x * blockDim.x + threadIdx.x;
  if (i < (long)N_NODES * HID_DIM) {
    int c = (int)(i & 127);
    float v = h[i] * scale[c] + shift[c];
    h[i] = fmaxf(v, 0.0f);
    agg2[i] = 0.0f;
  }
}

// ---------------- layer 2 GEMM: (agg2/cnt)@W2l + h@W2r + b2 -> out ---------
__global__ void k_gemm2(const float* __restrict__ agg2, const float* __restrict__ cnt,
                        const float* __restrict__ h,
                        const float* __restrict__ Wl, const float* __restrict__ Wr,
                        const float* __restrict__ bias,
                        float* __restrict__ out) {
  const int wave = blockIdx.x * (blockDim.x >> 5) + (threadIdx.x >> 5);
  if (wave >= N_NODES / 16) return;
  const int lane  = threadIdx.x & 31;
  const int laneM = lane & 15;
  const int kHalf = lane >> 4;
  const int m0 = wave * 16;
  const int rowA = m0 + laneM;

  const float rInv = 1.0f / fmaxf(cnt[rowA], 1.0f);
  const float* aRow = agg2 + (long)rowA * 128;
  const float* hRow = h    + (long)rowA * 128;

  v8f acc[4] = {};
  for (int k0 = 0; k0 < 128; k0 += 4) {
    const int kb = k0 + 2 * kHalf;
    v2f a1, a2;
    a1.x = aRow[kb] * rInv;  a1.y = aRow[kb + 1] * rInv;
    a2.x = hRow[kb];         a2.y = hRow[kb + 1];
#pragma unroll
    for (int t = 0; t < 4; ++t) {
      const int n = t * 16 + laneM;
      v2f bl, br;
      bl.x = Wl[kb * 64 + n];        bl.y = Wl[(kb + 1) * 64 + n];
      br.x = Wr[kb * 64 + n];        br.y = Wr[(kb + 1) * 64 + n];
      acc[t] = __builtin_amdgcn_wmma_f32_16x16x4_f32(false, a1, false, bl,
                                                     (short)0, acc[t], false, false);
      acc[t] = __builtin_amdgcn_wmma_f32_16x16x4_f32(false, a2, false, br,
                                                     (short)0, acc[t], false, false);
    }
  }
#pragma unroll
  for (int t = 0; t < 4; ++t) {
    const int n = t * 16 + laneM;
    const float bv = bias[n];
#pragma unroll
    for (int i = 0; i < 8; ++i) {
      const int row = m0 + i + (kHalf ? 8 : 0);
      out[(long)row * 64 + n] = acc[t][i] + bv;
    }
  }
}

// ---------------------------------------------------------------------------
extern "C" void kernel_launch(void* const* d_in, const int* in_sizes, int n_in,
                              void* d_out, int out_size, void* d_ws, size_t ws_size,
                              hipStream_t stream) {
  const float* x   = (const float*)d_in[0];
  const int*   ei  = (const int*)d_in[1];   // (2, E): row 0 = src, row 1 = dst
  const float* W1l = (const float*)d_in[2];
  const float* W1r = (const float*)d_in[3];
  const float* b1  = (const float*)d_in[4];
  const float* g1  = (const float*)d_in[5];
  const float* be1 = (const float*)d_in[6];
  const float* W2l = (const float*)d_in[7];
  const float* W2r = (const float*)d_in[8];
  const float* b2  = (const float*)d_in[9];
  float* out = (float*)d_out;

  // workspace layout (floats): aggA | cnt | bn(4x128) | h
  float* ws      = (float*)d_ws;
  float* aggA    = ws;                               // N_NODES*128
  float* cnt     = aggA + (long)N_NODES * 128;       // N_NODES
  float* bnSum   = cnt + N_NODES;                    // 128
  float* bnSq    = bnSum + 128;                      // 128
  float* bnScale = bnSq + 128;                       // 128
  float* bnShift = bnScale + 128;                    // 128
  float* hbuf    = bnShift + 128;                    // N_NODES*128

  const long zeroN = (long)N_NODES * 128 + N_NODES + 512;
  k_zero<<<2048, 256, 0, stream>>>(ws, zeroN);

  k_count<<<(N_EDGES + 255) / 256, 256, 0, stream>>>(ei, cnt);

  const int scatterBlocks = (int)(((long)N_EDGES * 32 + 255) / 256);
  k_scatter<<<scatterBlocks, 256, 0, stream>>>(ei, x, aggA);

  const int gemmBlocks = (N_NODES / 16 + 3) / 4;     // 4 waves (128 thr) / block
  k_gemm1<<<gemmBlocks, 128, 0, stream>>>(aggA, cnt, x, W1l, W1r, b1,
                                          hbuf, bnSum, bnSq);

  k_bnfin<<<1, 128, 0, stream>>>(bnSum, bnSq, g1, be1, bnScale, bnShift);

  k_bnrelu_zero<<<(int)(((long)N_NODES * 128 + 255) / 256), 256, 0, stream>>>(
      hbuf, aggA, bnScale, bnShift);

  k_scatter<<<scatterBlocks, 256, 0, stream>>>(ei, hbuf, aggA);

  k_gemm2<<<gemmBlocks, 128, 0, stream>>>(aggA, cnt, hbuf, W2l, W2r, b2, out);
}